// PointNetSetAbstraction_33621003993619
// MI455X (gfx1250) — compile-verified
//
#include <hip/hip_runtime.h>

#define BATCH   4
#define NPTS    8192
#define NPOINT  1024
#define NSAMPLE 32
#define CIN0    16
#define RADIUS2 0.04f
#define EPSV    1e-5f
#define ROWS    (BATCH * NPOINT * NSAMPLE)   // 131072

typedef __attribute__((ext_vector_type(2))) float v2f;
typedef __attribute__((ext_vector_type(8))) float v8f;

// ---------------------------------------------------------------- FPS
__device__ __forceinline__ void argmax_wave(float& v, int& i) {
#pragma unroll
  for (int off = 16; off > 0; off >>= 1) {
    float ov = __shfl_xor(v, off, 32);
    int   oi = __shfl_xor(i, off, 32);
    if (ov > v) { v = ov; i = oi; }
  }
}

__global__ __launch_bounds__(1024) void fps_kernel(const float* __restrict__ xyz,
                                                   int* __restrict__ fps_idx) {
  const int b = blockIdx.x;
  const float* x = xyz + (size_t)b * NPTS * 3;
  const int t = threadIdx.x;
  const int lane = t & 31, wid = t >> 5;
  __shared__ float wval[32];
  __shared__ int   widx[32];
  __shared__ int   sFar;

  float px[8], py[8], pz[8], dist[8];
#pragma unroll
  for (int i = 0; i < 8; ++i) {
    int j = t + i * 1024;
    px[i] = x[j * 3 + 0]; py[i] = x[j * 3 + 1]; pz[i] = x[j * 3 + 2];
    dist[i] = 1e10f;
  }
  int farIdx = 0;
  for (int s = 0; s < NPOINT; ++s) {
    if (t == 0) fps_idx[b * NPOINT + s] = farIdx;
    float cx = x[farIdx * 3 + 0], cy = x[farIdx * 3 + 1], cz = x[farIdx * 3 + 2];
    float best = -1.0f; int bj = 0;
#pragma unroll
    for (int i = 0; i < 8; ++i) {
      float dx = px[i] - cx, dy = py[i] - cy, dz = pz[i] - cz;
      float d = dx * dx + dy * dy + dz * dz;
      dist[i] = fminf(dist[i], d);
      if (dist[i] > best) { best = dist[i]; bj = t + i * 1024; }
    }
    argmax_wave(best, bj);
    if (lane == 0) { wval[wid] = best; widx[wid] = bj; }
    __syncthreads();
    if (wid == 0) {
      float v = wval[lane]; int j2 = widx[lane];
      argmax_wave(v, j2);
      if (lane == 0) sFar = j2;
    }
    __syncthreads();
    farIdx = sFar;
  }
}

// ---------------------------------------------------------------- gather new_xyz
__global__ void gather_newxyz(const float* __restrict__ xyz, const int* __restrict__ fps_idx,
                              float* __restrict__ out) {
  int i = blockIdx.x * blockDim.x + threadIdx.x;
  if (i >= BATCH * NPOINT) return;
  int b = i / NPOINT;
  int j = fps_idx[i];
  const float* p = xyz + ((size_t)b * NPTS + j) * 3;
  out[i * 3 + 0] = p[0]; out[i * 3 + 1] = p[1]; out[i * 3 + 2] = p[2];
}

// ---------------------------------------------------------------- ball query (wave per center)
__global__ __launch_bounds__(256) void ballq_kernel(const float* __restrict__ xyz,
                                                    const float* __restrict__ newxyz,
                                                    int* __restrict__ gidx) {
  int gt = blockIdx.x * blockDim.x + threadIdx.x;
  int w = gt >> 5;                    // center id, 0..4095
  int lane = threadIdx.x & 31;
  int b = w / NPOINT;
  const float* x = xyz + (size_t)b * NPTS * 3;
  float cx = newxyz[w * 3 + 0], cy = newxyz[w * 3 + 1], cz = newxyz[w * 3 + 2];
  int* out = gidx + (size_t)w * NSAMPLE;
  int cnt = 0;
  int firstIdx = 0;
  bool haveFirst = false;
  for (int base = 0; base < NPTS && cnt < NSAMPLE; base += 32) {
    int j = base + lane;
    float dx = x[j * 3 + 0] - cx, dy = x[j * 3 + 1] - cy, dz = x[j * 3 + 2] - cz;
    bool inside = (dx * dx + dy * dy + dz * dz) <= RADIUS2;
    unsigned mask = (unsigned)__ballot(inside);
    if (!haveFirst && mask) { firstIdx = base + __builtin_ctz(mask); haveFirst = true; }
    if (inside) {
      int pos = cnt + __popc(mask & ((1u << lane) - 1u));
      if (pos < NSAMPLE) out[pos] = j;
    }
    cnt += __popc(mask);
  }
  if (cnt < NSAMPLE) {
    for (int slot = cnt + lane; slot < NSAMPLE; slot += 32) out[slot] = firstIdx;
  }
}

// ---------------------------------------------------------------- grouped feature build (pad 19 -> 32)
__global__ void build_feat(const float* __restrict__ xyz, const float* __restrict__ points,
                           const float* __restrict__ newxyz, const int* __restrict__ gidx,
                           float* __restrict__ feat) {
  int r = blockIdx.x * blockDim.x + threadIdx.x;
  if (r >= ROWS) return;
  int k  = r & (NSAMPLE - 1);
  int bs = r >> 5;
  int b  = bs / NPOINT;
  int j  = gidx[bs * NSAMPLE + k];
  const float* p = xyz + ((size_t)b * NPTS + j) * 3;
  const float* c = newxyz + (size_t)bs * 3;
  float* f = feat + (size_t)r * 32;
  f[0] = p[0] - c[0]; f[1] = p[1] - c[1]; f[2] = p[2] - c[2];
  const float* pt = points + ((size_t)b * NPTS + j) * CIN0;
#pragma unroll
  for (int q = 0; q < CIN0; ++q) f[3 + q] = pt[q];
#pragma unroll
  for (int q = 19; q < 32; ++q) f[q] = 0.f;
}

// ---------------------------------------------------------------- WMMA GEMM + BN-stat accumulation
// Y[row, n] = sum_c act(X[row, c]) * W[n, c]  (ACT = fused BN+ReLU of previous layer)
// All shapes compile-time: fully unrolled k-loop and n-tile loop, weights staged
// zero-padded in LDS with +2 row pad (16 lanes -> 16 distinct banks on ds_load_b64).
template <int CIN, int CINPAD, int COUT, bool ACT>
__global__ __launch_bounds__(256) void gemm_bn(const float* __restrict__ X,
                                               const float* __restrict__ W,
                                               float* __restrict__ Y,
                                               const float* __restrict__ inScale,
                                               const float* __restrict__ inShift,
                                               float* __restrict__ stats) {
  constexpr int LSTRIDE = CINPAD + 2;
  constexpr int NT = COUT / 16;
  __shared__ float wlds[COUT * LSTRIDE];
  __shared__ float ssc[ACT ? CINPAD : 1];
  __shared__ float ssh[ACT ? CINPAD : 1];
  __shared__ float lsum[COUT];
  __shared__ float lsq[COUT];

  const int tid = threadIdx.x;
  for (int idx = tid; idx < COUT * CINPAD; idx += 256) {
    int n = idx / CINPAD, c = idx - n * CINPAD;
    wlds[n * LSTRIDE + c] = (c < CIN) ? W[n * CIN + c] : 0.f;
  }
  if (ACT) {
    for (int c = tid; c < CINPAD; c += 256) { ssc[c] = inScale[c]; ssh[c] = inShift[c]; }
  }
  for (int c = tid; c < COUT; c += 256) { lsum[c] = 0.f; lsq[c] = 0.f; }
  __syncthreads();

  const int wave  = tid >> 5, lane = tid & 31;
  const int hf    = lane >> 4, lm = lane & 15;
  const int mtile = blockIdx.x * 8 + wave;
  const float* xr = X + (size_t)(mtile * 16 + lm) * CINPAD;

  v8f acc[NT] = {};
#pragma unroll
  for (int k0 = 0; k0 < CINPAD; k0 += 4) {
    const int c0 = k0 + hf * 2;
    v2f a = *(const v2f*)(xr + c0);                 // aligned global_load_b64
    if (ACT) {
      a[0] = fmaxf(0.f, a[0] * ssc[c0]     + ssh[c0]);
      a[1] = fmaxf(0.f, a[1] * ssc[c0 + 1] + ssh[c0 + 1]);
    }
#pragma unroll
    for (int nt = 0; nt < NT; ++nt) {
      v2f bv = *(const v2f*)(&wlds[(nt * 16 + lm) * LSTRIDE + c0]);  // ds_load_b64
      acc[nt] = __builtin_amdgcn_wmma_f32_16x16x4_f32(
          false, a, false, bv, (short)0, acc[nt], false, false);
    }
  }

#pragma unroll
  for (int nt = 0; nt < NT; ++nt) {
    const int n = nt * 16 + lm;
    float s = 0.f, q = 0.f;
#pragma unroll
    for (int v = 0; v < 8; ++v) {
      const int row = mtile * 16 + hf * 8 + v;
      float y = acc[nt][v];
      Y[(size_t)row * COUT + n] = y;
      s += y; q += y * y;
    }
    atomicAdd(&lsum[n], s);
    atomicAdd(&lsq[n], q);
  }
  __syncthreads();
  for (int c = tid; c < COUT; c += 256) {
    atomicAdd(&stats[c],        lsum[c]);
    atomicAdd(&stats[COUT + c], lsq[c]);
  }
}

// ---------------------------------------------------------------- stats -> fused scale/shift
__global__ void bnparams_kernel(const float* __restrict__ stats, const float* __restrict__ g,
                                const float* __restrict__ be, float* __restrict__ sc,
                                float* __restrict__ sh, int Cout, float invR) {
  int c = threadIdx.x;
  if (c >= Cout) return;
  float mu  = stats[c] * invR;
  float var = stats[Cout + c] * invR - mu * mu;
  float s   = g[c] * rsqrtf(var + EPSV);
  sc[c] = s;
  sh[c] = be[c] - mu * s;
}

// ---------------------------------------------------------------- max over nsample (with fused BN+ReLU)
__global__ void maxpool_kernel(const float* __restrict__ Y3, const float* __restrict__ sc,
                               const float* __restrict__ sh, float* __restrict__ out) {
  int i = blockIdx.x * blockDim.x + threadIdx.x;
  if (i >= BATCH * NPOINT * 128) return;
  int o  = i & 127;
  int bs = i >> 7;
  float s = sc[o], h = sh[o];
  const float* y = Y3 + (size_t)bs * NSAMPLE * 128 + o;
  float m = 0.f;  // relu outputs are >= 0
#pragma unroll
  for (int k = 0; k < NSAMPLE; ++k)
    m = fmaxf(m, fmaxf(0.f, y[(size_t)k * 128] * s + h));
  out[i] = m;
}

// ---------------------------------------------------------------- host
extern "C" void kernel_launch(void* const* d_in, const int* in_sizes, int n_in,
                              void* d_out, int out_size, void* d_ws, size_t ws_size,
                              hipStream_t stream) {
  const float* xyz    = (const float*)d_in[0];
  const float* points = (const float*)d_in[1];
  const float* w1  = (const float*)d_in[2];
  const float* g1  = (const float*)d_in[4];
  const float* be1 = (const float*)d_in[5];
  const float* w2  = (const float*)d_in[6];
  const float* g2  = (const float*)d_in[8];
  const float* be2 = (const float*)d_in[9];
  const float* w3  = (const float*)d_in[10];
  const float* g3  = (const float*)d_in[12];
  const float* be3 = (const float*)d_in[13];
  float* out = (float*)d_out;

  char*  ws      = (char*)d_ws;
  int*   fps_idx = (int*)(ws + 0);                       // 16 KB
  int*   gidx    = (int*)(ws + 16384);                   // 512 KB
  float* stats   = (float*)(ws + 16384 + 524288);        // 3 * 256 floats
  float* bnsc    = stats + 3 * 256;                      // 3 * 256 floats (scale|shift)
  float* S1 = (float*)(ws + (size_t)(1 << 20));                       // 64 MB slot A
  float* S2 = (float*)(ws + (size_t)(1 << 20) + ((size_t)64 << 20));  // 64 MB slot B

  float* feat = S1;   // 131072 x 32
  float* Y1   = S2;   // 131072 x 64
  float* Y2   = S1;   // 131072 x 64  (feat dead)
  float* Y3   = S2;   // 131072 x 128 (Y1 dead)

  const float invR = 1.0f / (float)ROWS;

  fps_kernel<<<BATCH, 1024, 0, stream>>>(xyz, fps_idx);
  gather_newxyz<<<(BATCH * NPOINT + 255) / 256, 256, 0, stream>>>(xyz, fps_idx, out);
  ballq_kernel<<<(BATCH * NPOINT) / 8, 256, 0, stream>>>(xyz, out, gidx);
  build_feat<<<(ROWS + 255) / 256, 256, 0, stream>>>(xyz, points, out, gidx, feat);

  hipMemsetAsync(stats, 0, 3 * 256 * sizeof(float), stream);

  gemm_bn<19, 32, 64, false><<<ROWS / 128, 256, 0, stream>>>(
      feat, w1, Y1, nullptr, nullptr, stats);
  bnparams_kernel<<<1, 128, 0, stream>>>(stats, g1, be1, bnsc, bnsc + 128, 64, invR);

  gemm_bn<64, 64, 64, true><<<ROWS / 128, 256, 0, stream>>>(
      Y1, w2, Y2, bnsc, bnsc + 128, stats + 256);
  bnparams_kernel<<<1, 128, 0, stream>>>(stats + 256, g2, be2, bnsc + 256, bnsc + 384, 64, invR);

  gemm_bn<64, 64, 128, true><<<ROWS / 128, 256, 0, stream>>>(
      Y2, w3, Y3, bnsc + 256, bnsc + 384, stats + 512);
  bnparams_kernel<<<1, 128, 0, stream>>>(stats + 512, g3, be3, bnsc + 512, bnsc + 640, 128, invR);

  maxpool_kernel<<<(BATCH * NPOINT * 128 + 255) / 256, 256, 0, stream>>>(
      Y3, bnsc + 512, bnsc + 640, out + BATCH * NPOINT * 3);
}